// ClassBalancedSupConLoss_26061861552544
// MI455X (gfx1250) — compile-verified
//
#include <hip/hip_runtime.h>
#include <hip/hip_bf16.h>

typedef __attribute__((ext_vector_type(2))) float v2f;
typedef __attribute__((ext_vector_type(8))) float v8f;

constexpr int   D_DIM     = 128;   // embedding dim
constexpr int   CCH       = 64;    // columns staged per LDS chunk (= 4 col tiles)
constexpr int   LSTRIDE   = 132;   // padded floats per column in LDS (conflict-free b64 reads)
constexpr float BASE_TEMP = 0.07f;
constexpr float EPS_C     = 1e-8f;

__device__ __forceinline__ float cls_invt(int c) {
    // CLASS_TEMPS = {0.08, 0.05, 0.10} -> 1/t
    return c == 0 ? 12.5f : (c == 1 ? 20.0f : 10.0f);
}

// ---------------- kernel 1: class histogram (3 bins, deterministic int atomics) ----
__global__ void supcon_count(const int* __restrict__ labels, int* __restrict__ counts, int n) {
    __shared__ int sc[3];
    if (threadIdx.x < 3) sc[threadIdx.x] = 0;
    __syncthreads();
    for (int i = threadIdx.x; i < n; i += blockDim.x) {
        int l = labels[i];
        if (l >= 0 && l < 3) atomicAdd(&sc[l], 1);
    }
    __syncthreads();
    if (threadIdx.x < 3) counts[threadIdx.x] = sc[threadIdx.x];
}

// ---------------- kernel 2: fused E*E^T + softmax stats + per-row loss -------------
// block = 128 threads (4 waves). Each wave owns one 16-row tile -> block covers 64 rows.
__global__ void __launch_bounds__(128)
supcon_main(const float* __restrict__ emb, const int* __restrict__ labels,
            const int* __restrict__ counts, float* __restrict__ ws_loss, int Bn) {
    __shared__ float ldsB[CCH * LSTRIDE];
    __shared__ int   ldsLab[CCH];

    const int tid  = threadIdx.x;
    const int wave = tid >> 5;
    const int lane = tid & 31;
    const int half = lane >> 4;      // 0: lanes 0-15, 1: lanes 16-31
    const int l15  = lane & 15;
    const int M0   = (blockIdx.x * 4 + wave) * 16;

    // ---- A fragments: 16x128 fp32 tile, WMMA 16x16x4 layout, 64 floats per lane ----
    // lane<16 carries K = 4k,4k+1 of row M0+l15 ; lane>=16 carries K = 4k+2,4k+3
    const float* arow_ptr = emb + (long)(M0 + l15) * D_DIM + half * 2;
    float a[64];
#pragma unroll
    for (int kk = 0; kk < 32; ++kk) {
        v2f t = *(const v2f*)(arow_ptr + kk * 4);
        a[2 * kk]     = t.x;
        a[2 * kk + 1] = t.y;
    }

    // ---- |e_row|^2 from the resident fragment: local half-sum + cross-half swap ----
    float nloc = 0.0f;
#pragma unroll
    for (int i = 0; i < 64; ++i) nloc = fmaf(a[i], a[i], nloc);
    const float nfull = nloc + __shfl_xor(nloc, 16, 32);   // lane L: |e_{M0+(L&15)}|^2

    // per-(row,lane) state: VGPR r holds row M0+r (lanes 0-15) / M0+r+8 (lanes 16-31)
    int   rowlab[8];
    float invt[8], n2[8], pz[8], pp[8];
#pragma unroll
    for (int r = 0; r < 8; ++r) {
        int row   = M0 + r + 8 * half;
        rowlab[r] = labels[row];
        invt[r]   = cls_invt(rowlab[r]);
        n2[r]     = __shfl(nfull, r + 8 * half, 32);       // |e_row|^2 (fixed softmax offset)
        pz[r]     = 0.0f;
        pp[r]     = 0.0f;
    }

    for (int cb = 0; cb < Bn; cb += CCH) {
        __syncthreads();   // previous chunk fully consumed
        // cooperative stage: 64 cols x 128 floats = 2048 float4, 16 per thread
#pragma unroll
        for (int it = 0; it < (CCH * D_DIM / 4) / 128; ++it) {
            int    idx = it * 128 + tid;
            int    col = idx >> 5;        // / (D/4)
            int    k4  = idx & 31;        // % (D/4)
            float4 v   = ((const float4*)(emb + (long)(cb + col) * D_DIM))[k4];
            float* dst = &ldsB[col * LSTRIDE + k4 * 4];
            dst[0] = v.x; dst[1] = v.y; dst[2] = v.z; dst[3] = v.w;
        }
        if (tid < CCH) ldsLab[tid] = labels[cb + tid];
        if (cb + CCH < Bn) __builtin_prefetch(emb + (long)(cb + CCH) * D_DIM, 0, 1);
        __syncthreads();

        // all 4 column tiles of the chunk concurrently -> 4 independent WMMA chains
        const float* b0 = &ldsB[l15 * LSTRIDE + half * 2];
        const float* b1 = b0 + 16 * LSTRIDE;
        const float* b2 = b0 + 32 * LSTRIDE;
        const float* b3 = b0 + 48 * LSTRIDE;
        v8f c0 = {}, c1 = {}, c2 = {}, c3 = {};
#pragma unroll
        for (int kk = 0; kk < 32; ++kk) {
            v2f av  = {a[2 * kk], a[2 * kk + 1]};
            v2f bv0 = *(const v2f*)(b0 + kk * 4);
            v2f bv1 = *(const v2f*)(b1 + kk * 4);
            v2f bv2 = *(const v2f*)(b2 + kk * 4);
            v2f bv3 = *(const v2f*)(b3 + kk * 4);
            c0 = __builtin_amdgcn_wmma_f32_16x16x4_f32(false, av, false, bv0,
                                                       (short)0, c0, false, false);
            c1 = __builtin_amdgcn_wmma_f32_16x16x4_f32(false, av, false, bv1,
                                                       (short)0, c1, false, false);
            c2 = __builtin_amdgcn_wmma_f32_16x16x4_f32(false, av, false, bv2,
                                                       (short)0, c2, false, false);
            c3 = __builtin_amdgcn_wmma_f32_16x16x4_f32(false, av, false, bv3,
                                                       (short)0, c3, false, false);
        }
        const int colLab0 = ldsLab[l15];
        const int colLab1 = ldsLab[16 + l15];
        const int colLab2 = ldsLab[32 + l15];
        const int colLab3 = ldsLab[48 + l15];
        const int col0    = cb + l15;
#pragma unroll
        for (int r = 0; r < 8; ++r) {
            const int row = M0 + r + 8 * half;
            // fixed-offset softmax accumulation: exp((sim - |e_row|^2) / t_row)
            pz[r] += __expf((c0[r] - n2[r]) * invt[r]);
            pz[r] += __expf((c1[r] - n2[r]) * invt[r]);
            pz[r] += __expf((c2[r] - n2[r]) * invt[r]);
            pz[r] += __expf((c3[r] - n2[r]) * invt[r]);
            if ((colLab0 == rowlab[r]) && (col0      != row)) pp[r] += c0[r];
            if ((colLab1 == rowlab[r]) && (col0 + 16 != row)) pp[r] += c1[r];
            if ((colLab2 == rowlab[r]) && (col0 + 32 != row)) pp[r] += c2[r];
            if ((colLab3 == rowlab[r]) && (col0 + 48 != row)) pp[r] += c3[r];
        }
    }

    // ---- merge the 16 lane-partials per row (xor masks stay inside each half) ----
#pragma unroll
    for (int r = 0; r < 8; ++r) {
#pragma unroll
        for (int off = 8; off >= 1; off >>= 1) {
            pz[r] += __shfl_xor(pz[r], off, 32);
            pp[r] += __shfl_xor(pp[r], off, 32);
        }
    }

    if (l15 == 0) {   // lanes 0 and 16 finalize rows r and r+8
#pragma unroll
        for (int r = 0; r < 8; ++r) {
            int   row = M0 + r + 8 * half;
            int   cnt = counts[rowlab[r]] - 1;     // positives excluding self
            float loss = 0.0f;
            if (cnt > 0) {
                float fc = (float)cnt;
                float m0 = n2[r] * invt[r];        // fixed offset (== diag logit)
                float lp = (pp[r] * invt[r] - fc * m0) - fc * __logf(pz[r] + EPS_C);
                loss = -(BASE_TEMP * invt[r]) * lp / (fc + EPS_C);
            }
            ws_loss[row] = loss;
        }
    }
}

// ---------------- kernel 3: deterministic fixed-order final reduction --------------
__global__ void supcon_reduce(const float* __restrict__ ws_loss, const int* __restrict__ counts,
                              float* __restrict__ out, int n) {
    __shared__ float sh[256];
    float acc = 0.0f;
    for (int i = threadIdx.x; i < n; i += 256) acc += ws_loss[i];
    sh[threadIdx.x] = acc;
    __syncthreads();
    for (int s = 128; s > 0; s >>= 1) {
        if ((int)threadIdx.x < s) sh[threadIdx.x] += sh[threadIdx.x + s];
        __syncthreads();
    }
    if (threadIdx.x == 0) {
        float nv = 0.0f;
        for (int c = 0; c < 3; ++c) {
            int cc = counts[c];
            if (cc >= 2) nv += (float)cc;   // rows of class c are valid iff count>=2
        }
        out[0] = (nv > 0.0f) ? sh[0] / fmaxf(nv, 1.0f) : 0.0f;
    }
}

extern "C" void kernel_launch(void* const* d_in, const int* in_sizes, int n_in,
                              void* d_out, int out_size, void* d_ws, size_t ws_size,
                              hipStream_t stream) {
    const float* emb    = (const float*)d_in[0];
    const int*   labels = (const int*)d_in[1];
    const int    Bn     = in_sizes[1];          // 8192

    float* ws_loss = (float*)d_ws;              // Bn floats
    int*   counts  = (int*)(ws_loss + Bn);      // 3 ints

    supcon_count<<<1, 256, 0, stream>>>(labels, counts, Bn);
    supcon_main<<<Bn / 64, 128, 0, stream>>>(emb, labels, counts, ws_loss, Bn);
    supcon_reduce<<<1, 256, 0, stream>>>(ws_loss, counts, (float*)d_out, Bn);
}